// HMCLayer_28716151341162
// MI455X (gfx1250) — compile-verified
//
#include <hip/hip_runtime.h>
#include <hip/hip_bf16.h>

typedef __attribute__((ext_vector_type(2))) float v2f;
typedef __attribute__((ext_vector_type(8))) float v8f;

#define LSLOPE 0.2f

static constexpr int cN0 = 100000, cN1 = 300000, cN2 = 200000;
static constexpr int cNNZ_A0 = 600000, cNNZ_A1 = 1200000, cNNZ_C2 = 600000;
static constexpr int cNNZ_B1 = 600000, cNNZ_B2 = 600000;
static constexpr int cNSEG_MAX = 300000;   // max(N0,N1,N2)
static constexpr int cNNZ_MAX = 1200000;

// LDS layout for pre-paired W: 32 pair-rows, stride 80 float2 (bank-conflict-free)
#define WPAIR_STRIDE 80

// ---------------------------------------------------------------- fills
__global__ void fill_f32_k(float* __restrict__ p, float v, int n) {
  int i = blockIdx.x * blockDim.x + threadIdx.x;
  int stride = gridDim.x * blockDim.x;
  for (; i < n; i += stride) p[i] = v;
}
__global__ void fill_u32_k(unsigned* __restrict__ p, unsigned v, int n) {
  int i = blockIdx.x * blockDim.x + threadIdx.x;
  int stride = gridDim.x * blockDim.x;
  for (; i < n; i += stride) p[i] = v;
}

// ---------------------------------------------------------------- GEMM: M = X @ W
// X: [N,64] row-major, W: [64,64] row-major, M: [N,64].
// One wave -> one 16-row x 64-col tile. fp32 WMMA 16x16x4, K loop of 16 steps.
// W staged in LDS pre-paired: lW2[j*WPAIR_STRIDE + n] = (W[2j][n], W[2j+1][n])
// so every B fragment is a single aligned ds_load_b64.
__global__ __launch_bounds__(256)
void gemm_xw_wmma(const float* __restrict__ X, const float* __restrict__ W,
                  float* __restrict__ M, int N) {
  __shared__ v2f lW2[32 * WPAIR_STRIDE];
  for (int i = threadIdx.x; i < 32 * 64; i += 256) {
    const int j = i >> 6;          // pair-row 0..31  (K = 2j, 2j+1)
    const int n = i & 63;          // column
    v2f p;
    p.x = W[(2 * j) * 64 + n];
    p.y = W[(2 * j + 1) * 64 + n];
    lW2[j * WPAIR_STRIDE + n] = p;
  }
  __syncthreads();

  const int lane  = threadIdx.x & 31;
  const int wid   = threadIdx.x >> 5;
  const int row0  = (blockIdx.x * 8 + wid) * 16;
  if (row0 >= N) return;                      // wave-uniform: EXEC stays all-1s

  const int nc    = lane & 15;                // col-within-16 / M-within-16
  const int khalf = lane >> 4;                // 0: K=k0,k0+1  1: K=k0+2,k0+3
  const float* xrow = X + (size_t)(row0 + nc) * 64;

  v8f acc0 = {}, acc1 = {}, acc2 = {}, acc3 = {};
#pragma unroll
  for (int k0 = 0; k0 < 64; k0 += 4) {
    const int kk  = k0 + 2 * khalf;           // first K of this lane's pair
    const int kkh = (k0 >> 1) + khalf;        // pair-row index in lW2
    v2f a;  a.x = xrow[kk];  a.y = xrow[kk + 1];
    const v2f* bb = &lW2[kkh * WPAIR_STRIDE + nc];
    v2f b0 = bb[0];
    v2f b1 = bb[16];
    v2f b2 = bb[32];
    v2f b3 = bb[48];
    acc0 = __builtin_amdgcn_wmma_f32_16x16x4_f32(false, a, false, b0, (short)0, acc0, false, false);
    acc1 = __builtin_amdgcn_wmma_f32_16x16x4_f32(false, a, false, b1, (short)0, acc1, false, false);
    acc2 = __builtin_amdgcn_wmma_f32_16x16x4_f32(false, a, false, b2, (short)0, acc2, false, false);
    acc3 = __builtin_amdgcn_wmma_f32_16x16x4_f32(false, a, false, b3, (short)0, acc3, false, false);
  }

  const int rbase = row0 + 8 * khalf;         // D: VGPR r -> M=r (lanes0-15) / r+8 (16-31)
#pragma unroll
  for (int r = 0; r < 8; ++r) {
    float* orow = M + (size_t)(rbase + r) * 64 + nc;
    orow[0]  = acc0[r];
    orow[16] = acc1[r];
    orow[32] = acc2[r];
    orow[48] = acc3[r];
  }
}

// ---------------------------------------------------------------- per-row logits
// lLo[r] = m[r]·a[0:64], lHi[r] = m[r]·a[64:128]. One wave per row.
__global__ __launch_bounds__(256)
void row_logits_k(const float* __restrict__ M, const float* __restrict__ a,
                  float* __restrict__ lLo, float* __restrict__ lHi, int N) {
  const int lane = threadIdx.x & 31;
  const int row  = (blockIdx.x * blockDim.x + threadIdx.x) >> 5;
  if (row >= N) return;
  const float* mr = M + (size_t)row * 64;
  const float v0 = mr[lane], v1 = mr[32 + lane];
  float p = v0 * a[lane]      + v1 * a[32 + lane];
  float q = v0 * a[64 + lane] + v1 * a[96 + lane];
#pragma unroll
  for (int off = 16; off; off >>= 1) {
    p += __shfl_xor(p, off, 32);
    q += __shfl_xor(q, off, 32);
  }
  if (lane == 0) { lLo[row] = p; lHi[row] = q; }
}

// ---------------------------------------------------------------- segment softmax
__device__ __forceinline__ unsigned fkey(float f) {
  unsigned u = __float_as_uint(f);
  return (u & 0x80000000u) ? ~u : (u | 0x80000000u);   // order-preserving key
}
__device__ __forceinline__ float fkey_inv(unsigned k) {
  unsigned u = (k & 0x80000000u) ? (k ^ 0x80000000u) : ~k;
  return __uint_as_float(u);
}

// g[k] = leaky(la[ia[k]] + lb[ib[k]]); smax[seg[k]] = max(...)
__global__ __launch_bounds__(256)
void edge_logit_max_k(const float* __restrict__ la, const int* __restrict__ ia,
                      const float* __restrict__ lb, const int* __restrict__ ib,
                      const int* __restrict__ seg, float* __restrict__ g,
                      unsigned* __restrict__ smax, int nnz) {
  int k = blockIdx.x * blockDim.x + threadIdx.x;
  if (k >= nnz) return;
  float x = la[ia[k]] + lb[ib[k]];
  x = x > 0.0f ? x : LSLOPE * x;
  g[k] = x;
  atomicMax(&smax[seg[k]], fkey(x));
}

// g[k] = exp(g[k] - max[seg]); ssum[seg] += g[k]
__global__ __launch_bounds__(256)
void edge_exp_sum_k(float* __restrict__ g, const int* __restrict__ seg,
                    const unsigned* __restrict__ smax, float* __restrict__ ssum,
                    int nnz) {
  int k = blockIdx.x * blockDim.x + threadIdx.x;
  if (k >= nnz) return;
  const int s = seg[k];
  const float e = expf(g[k] - fkey_inv(smax[s]));
  g[k] = e;
  atomicAdd(&ssum[s], e);
}

// one wave per edge: out[seg[k]] += (g[k]/ssum[seg[k]]) * M[src[k]]  (64 floats)
__global__ __launch_bounds__(256)
void edge_scatter_k(const float* __restrict__ g, const int* __restrict__ seg,
                    const int* __restrict__ src, const float* __restrict__ ssum,
                    const float* __restrict__ M, float* __restrict__ out, int nnz) {
  const int lane = threadIdx.x & 31;
  const int k = (blockIdx.x * blockDim.x + threadIdx.x) >> 5;
  if (k >= nnz) return;
  const int s = seg[k], c = src[k];
  const float w = g[k] / ssum[s];
  const float* mr = M + (size_t)c * 64;
  float* orow = out + (size_t)s * 64;
  atomicAdd(&orow[lane],      w * mr[lane]);
  atomicAdd(&orow[32 + lane], w * mr[32 + lane]);
}

// ---------------------------------------------------------------- host
extern "C" void kernel_launch(void* const* d_in, const int* in_sizes, int n_in,
                              void* d_out, int out_size, void* d_ws, size_t ws_size,
                              hipStream_t stream) {
  (void)in_sizes; (void)n_in; (void)out_size; (void)ws_size;

  const float* x0 = (const float*)d_in[0];
  const float* x1 = (const float*)d_in[1];
  const float* x2 = (const float*)d_in[2];
  const int* adj0_r   = (const int*)d_in[3];
  const int* adj0_c   = (const int*)d_in[4];
  const int* adj1_r   = (const int*)d_in[5];
  const int* adj1_c   = (const int*)d_in[6];
  const int* coadj2_r = (const int*)d_in[7];
  const int* coadj2_c = (const int*)d_in[8];
  const int* inc1_r   = (const int*)d_in[9];   // targets in N0
  const int* inc1_c   = (const int*)d_in[10];  // sources in N1
  const int* inc2_r   = (const int*)d_in[11];  // targets in N1
  const int* inc2_c   = (const int*)d_in[12];  // sources in N2
  const float* W_hbs0_l1 = (const float*)d_in[13];
  const float* W_hbs0_l2 = (const float*)d_in[14];
  const float* W_hbs1_l2 = (const float*)d_in[15];
  const float* W_hbs2_l2 = (const float*)d_in[16];
  const float* Ws01_l1 = (const float*)d_in[17];
  const float* Wt01_l1 = (const float*)d_in[18];
  const float* Ws12_l1 = (const float*)d_in[19];
  const float* Wt12_l1 = (const float*)d_in[20];
  const float* Ws01_l2 = (const float*)d_in[21];
  const float* Wt01_l2 = (const float*)d_in[22];
  const float* Ws12_l2 = (const float*)d_in[23];
  const float* Wt12_l2 = (const float*)d_in[24];
  const float* a_hbs0_l1 = (const float*)d_in[25];
  const float* a_hbs0_l2 = (const float*)d_in[26];
  const float* a_hbs1_l2 = (const float*)d_in[27];
  const float* a_hbs2_l2 = (const float*)d_in[28];
  const float* a01_l1 = (const float*)d_in[29];
  const float* a12_l1 = (const float*)d_in[30];
  const float* a01_l2 = (const float*)d_in[31];
  const float* a12_l2 = (const float*)d_in[32];

  // ---- workspace carve (floats) ----
  float* ws = (float*)d_ws;
  size_t off = 0;
  auto carve = [&](size_t n) { float* p = ws + off; off += n; return p; };
  float* XL1_0 = carve((size_t)cN0 * 64);
  float* XL1_1 = carve((size_t)cN1 * 64);
  float* XL1_2 = carve((size_t)cN2 * 64);
  float* M0 = carve((size_t)cN0 * 64);
  float* M1 = carve((size_t)cN1 * 64);
  float* M2 = carve((size_t)cN2 * 64);
  float* LLO0 = carve(cN0); float* LHI0 = carve(cN0);
  float* LLO1 = carve(cN1); float* LHI1 = carve(cN1);
  float* LLO2 = carve(cN2); float* LHI2 = carve(cN2);
  unsigned* SMAX_E = (unsigned*)carve(cNSEG_MAX);
  float*    SSUM_E = carve(cNSEG_MAX);
  unsigned* SMAX_F = (unsigned*)carve(cNSEG_MAX);
  float*    SSUM_F = carve(cNSEG_MAX);
  float* GE = carve(cNNZ_MAX);
  float* GF = carve(cNNZ_MAX);

  float* OUT0 = (float*)d_out;
  float* OUT1 = OUT0 + (size_t)cN0 * 64;
  float* OUT2 = OUT1 + (size_t)cN1 * 64;

  // ---- launch helpers ----
  auto fillf = [&](float* p, float v, int n) {
    int grid = (n + 255) / 256; if (grid > 4096) grid = 4096;
    fill_f32_k<<<grid, 256, 0, stream>>>(p, v, n);
  };
  auto fillu = [&](unsigned* p, unsigned v, int n) {
    int grid = (n + 255) / 256; if (grid > 4096) grid = 4096;
    fill_u32_k<<<grid, 256, 0, stream>>>(p, v, n);
  };
  auto gemm = [&](const float* X, const float* W, float* M, int N) {
    gemm_xw_wmma<<<(N + 127) / 128, 256, 0, stream>>>(X, W, M, N);
  };
  auto logits = [&](const float* M, const float* a, float* lLo, float* lHi, int N) {
    row_logits_k<<<(N + 7) / 8, 256, 0, stream>>>(M, a, lLo, lHi, N);
  };
  // segment-softmax + aggregation over edge list
  auto softmax_agg = [&](const float* la, const int* ia, const float* lb, const int* ib,
                         const int* seg, int nseg, const float* Msrc, const int* src,
                         float* outAcc, float* gbuf, unsigned* smax, float* ssum, int nnz) {
    fillu(smax, 0u, nseg);
    fillf(ssum, 0.0f, nseg);
    edge_logit_max_k<<<(nnz + 255) / 256, 256, 0, stream>>>(la, ia, lb, ib, seg, gbuf, smax, nnz);
    edge_exp_sum_k<<<(nnz + 255) / 256, 256, 0, stream>>>(gbuf, seg, smax, ssum, nnz);
    edge_scatter_k<<<(nnz + 7) / 8, 256, 0, stream>>>(gbuf, seg, src, ssum, Msrc, outAcc, nnz);
  };
  // same-rank attention block
  auto run_hbs = [&](const float* x, const int* rows, const int* cols,
                     const float* W, const float* a, int n, int nnz,
                     float* mbuf, float* lLo, float* lHi, float* outAcc) {
    gemm(x, W, mbuf, n);
    logits(mbuf, a, lLo, lHi, n);
    softmax_agg(lLo, rows, lHi, cols, rows, n, mbuf, cols, outAcc, GE, SMAX_E, SSUM_E, nnz);
  };
  // cross-rank attention block; out_t may be null (level 2 discards msg_t)
  auto run_hbns = [&](const float* xs, const float* xt,
                      const int* t_rows, const int* s_cols,
                      const float* Ws, const float* Wt, const float* a,
                      int n_s, int n_t, int nnz,
                      float* smbuf, float* tmbuf,
                      float* lLo_s, float* lHi_s, float* lLo_t, float* lHi_t,
                      float* out_s, float* out_t) {
    gemm(xs, Ws, smbuf, n_s);
    gemm(xt, Wt, tmbuf, n_t);
    logits(smbuf, a, lLo_s, lHi_s, n_s);
    logits(tmbuf, a, lLo_t, lHi_t, n_t);
    if (out_t) {  // e: leaky(smLo[s_cols] + tmHi[t_rows]); softmax over t_rows; msg_t += att*sm[s_cols]
      softmax_agg(lLo_s, s_cols, lHi_t, t_rows, t_rows, n_t,
                  smbuf, s_cols, out_t, GE, SMAX_E, SSUM_E, nnz);
    }
    //  f: leaky(tmLo[t_rows] + smHi[s_cols]); softmax over s_cols; msg_s += att*tm[t_rows]
    softmax_agg(lLo_t, t_rows, lHi_s, s_cols, s_cols, n_s,
                tmbuf, t_rows, out_s, GF, SMAX_F, SSUM_F, nnz);
  };

  // ---- zero accumulators ----
  fillf(XL1_0, 0.0f, cN0 * 64);
  fillf(XL1_1, 0.0f, cN1 * 64);
  fillf(XL1_2, 0.0f, cN2 * 64);
  fillf(OUT0, 0.0f, (cN0 + cN1 + cN2) * 64);

  // ---- level 1 ----
  run_hbs(x0, adj0_r, adj0_c, W_hbs0_l1, a_hbs0_l1, cN0, cNNZ_A0, M0, LLO0, LHI0, XL1_0);
  run_hbns(x1, x0, inc1_r, inc1_c, Ws01_l1, Wt01_l1, a01_l1,
           cN1, cN0, cNNZ_B1, M1, M0, LLO1, LHI1, LLO0, LHI0,
           /*out_s=*/XL1_1, /*out_t=*/XL1_0);
  run_hbns(x2, x1, inc2_r, inc2_c, Ws12_l1, Wt12_l1, a12_l1,
           cN2, cN1, cNNZ_B2, M2, M1, LLO2, LHI2, LLO1, LHI1,
           /*out_s=*/XL1_2, /*out_t=*/XL1_1);

  // ---- level 2 ----
  run_hbs(XL1_0, adj0_r, adj0_c, W_hbs0_l2, a_hbs0_l2, cN0, cNNZ_A0, M0, LLO0, LHI0, OUT0);
  run_hbs(XL1_1, adj1_r, adj1_c, W_hbs1_l2, a_hbs1_l2, cN1, cNNZ_A1, M1, LLO1, LHI1, OUT1);
  run_hbs(XL1_2, coadj2_r, coadj2_c, W_hbs2_l2, a_hbs2_l2, cN2, cNNZ_C2, M2, LLO2, LHI2, OUT2);
  run_hbns(XL1_1, XL1_0, inc1_r, inc1_c, Ws01_l2, Wt01_l2, a01_l2,
           cN1, cN0, cNNZ_B1, M1, M0, LLO1, LHI1, LLO0, LHI0,
           /*out_s=*/OUT1, /*out_t=*/nullptr);
  run_hbns(XL1_2, XL1_1, inc2_r, inc2_c, Ws12_l2, Wt12_l2, a12_l2,
           cN2, cN1, cNNZ_B2, M2, M1, LLO2, LHI2, LLO1, LHI1,
           /*out_s=*/OUT2, /*out_t=*/nullptr);
}